// CapsNet_69114613730025
// MI455X (gfx1250) — compile-verified
//
#include <hip/hip_runtime.h>

typedef __attribute__((ext_vector_type(2))) float v2f;
typedef __attribute__((ext_vector_type(8))) float v8f;
typedef __attribute__((ext_vector_type(4))) unsigned int u32x4;
typedef __attribute__((ext_vector_type(8))) unsigned int u32x8;

// ---------------------------------------------------------------------------
// Tensor Data Mover: DMA a (rows x 32 f32) tile, row stride = row_stride_elems,
// from global memory into LDS at byte offset lds_off (contiguous fill).
// Descriptor per CDNA5 ISA ch.8 (D# group0 = 128b, group1 = 256b), issued with
// the 2-group form of TENSOR_LOAD_TO_LDS; completion tracked by TENSORcnt.
// ---------------------------------------------------------------------------
__device__ __forceinline__ void tdm_load_tile_to_lds(const float* gsrc,
                                                     unsigned lds_off,
                                                     unsigned rows,
                                                     unsigned row_stride_elems)
{
    const unsigned long long ga = (unsigned long long)(size_t)gsrc;
    u32x4 g0;
    g0[0] = 1u;                                    // count=1, user D#, not restore
    g0[1] = lds_off;                               // lds_addr (bytes)
    g0[2] = (unsigned)(ga & 0xffffffffu);          // global_addr[31:0]
    g0[3] = ((unsigned)(ga >> 32) & 0x01ffffffu)   // global_addr[56:32]
            | (2u << 30);                          // type = 2 ("image")
    u32x8 g1;
    g1[0] = (2u << 16);                            // data_size=2 (4B); mask/pad/iter=0
    g1[1] = (row_stride_elems & 0xffffu) << 16;    // tensor_dim0[15:0]
    g1[2] = (row_stride_elems >> 16)               // tensor_dim0[31:16]
            | ((rows & 0xffffu) << 16);            // tensor_dim1[15:0]
    g1[3] = (rows >> 16) | (32u << 16);            // tensor_dim1[31:16] | tile_dim0=32
    g1[4] = rows & 0xffffu;                        // tile_dim1 = rows, tile_dim2=0
    g1[5] = row_stride_elems;                      // tensor_dim0_stride[31:0]
    g1[6] = 0u;                                    // dim0_stride[47:32] | dim1_stride lo
    g1[7] = 0u;
    asm volatile("tensor_load_to_lds %0, %1" :: "s"(g0), "s"(g1) : "memory");
}

// ---------------------------------------------------------------------------
// conv1: 1->32, 5x5, stride 2, pad 2, ReLU. Output NHWC-flat h[b][(i*14+j)*32+c]
// ---------------------------------------------------------------------------
__global__ __launch_bounds__(256)
void conv1_relu_kernel(const float* __restrict__ x,
                       const float* __restrict__ w,
                       const float* __restrict__ bias,
                       float* __restrict__ out)
{
    __shared__ float xs[28 * 28];
    __shared__ float ws[32 * 25];
    __shared__ float bs[32];
    const int b = blockIdx.x;
    const float* xb = x + (size_t)b * 784;
    for (int i = threadIdx.x; i < 784; i += blockDim.x) xs[i] = xb[i];
    for (int i = threadIdx.x; i < 800; i += blockDim.x) ws[i] = w[i];
    if (threadIdx.x < 32) bs[threadIdx.x] = bias[threadIdx.x];
    __syncthreads();
    float* ob = out + (size_t)b * 6272;
    for (int n = threadIdx.x; n < 6272; n += blockDim.x) {
        const int c = n & 31;
        const int pix = n >> 5;
        const int i = pix / 14, j = pix % 14;
        float acc = bs[c];
#pragma unroll
        for (int kh = 0; kh < 5; ++kh) {
            const int ih = 2 * i - 2 + kh;
            if ((unsigned)ih >= 28u) continue;
#pragma unroll
            for (int kw = 0; kw < 5; ++kw) {
                const int iw = 2 * j - 2 + kw;
                if ((unsigned)iw >= 28u) continue;
                acc += ws[c * 25 + kh * 5 + kw] * xs[ih * 28 + iw];
            }
        }
        ob[n] = fmaxf(acc, 0.0f);
    }
}

// ---------------------------------------------------------------------------
// Grouped-conv-as-GEMM via V_WMMA_F32_16X16X4_F32, fully template-specialized.
//   Stage 1: TDM tensor_load_to_lds DMA of this group's 32-channel slice.
//   Stage 2: LDS->LDS im2col, padded to Npad columns (k-decode done ONCE).
//   Stage 3: per-wave 16x16 tiles; inner loop = b64 A load + 2 ds B loads + WMMA.
// Input read as  X[b*inBS + pix*Cin + (g*32 + r)]   (NHWC-flat / reinterpret)
// Weights (OC_total, 32, KK, KK) flat: row co contiguous length Ktot.
// Output: scale*(acc+bias); sigmoid for co < SIGCNT;
//         NCHW ? [co*PixOut+n] : [n*OC_total+co].
// ---------------------------------------------------------------------------
template <int Hin, int Win, int Hout, int Wout, int KK, int STRIDE,
          int Cin, int OCg, int OC_total, int SIGCNT, int NCHW>
__global__ __launch_bounds__(128)
void caps_conv_wmma_kernel(const float* __restrict__ X, const float* __restrict__ W,
                           const float* __restrict__ Bias, float* __restrict__ O,
                           long long inBS, long long outBS, float scale)
{
    constexpr int PixIn  = Hin * Win;
    constexpr int PixOut = Hout * Wout;
    constexpr int KK2    = KK * KK;
    constexpr int Ktot   = 32 * KK2;
    constexpr int Mt     = (OCg + 15) / 16;
    constexpr int Nt     = (PixOut + 15) / 16;
    constexpr int Npad   = Nt * 16;

    __shared__ float Xs[PixIn * 32];    // input slice, [pix][r]
    __shared__ float Xc[Ktot * Npad];   // im2col, [k][n] (pad cols unwritten/unused)

    const int b = blockIdx.x;
    const int g = blockIdx.y;
    const float* Xb = X + (size_t)b * inBS + g * 32;

    // Stage 1: slice -> LDS via the Tensor Data Mover (wave 0 issues; TDM
    // ignores EXEC, signals only the issuing wave, so wave 0 drains TENSORcnt
    // and the workgroup barrier publishes the LDS tile to all waves).
    if (threadIdx.x < 32) {
        tdm_load_tile_to_lds(Xb, (unsigned)(size_t)&Xs[0], PixIn, Cin);
        __builtin_amdgcn_s_wait_tensorcnt(0);
    }
    __syncthreads();

    // Stage 2: im2col into padded [k][n] layout
    if constexpr ((PixOut & 3) == 0) {
        for (int q = threadIdx.x; q < Ktot * (PixOut / 4); q += blockDim.x) {
            const int k  = q / (PixOut / 4);
            const int n4 = (q - k * (PixOut / 4)) * 4;
            const int r = k / KK2, tt = k - r * KK2;
            const int kh = tt / KK, kw = tt - kh * KK;
            float4 vv;
            float* pv = (float*)&vv;
#pragma unroll
            for (int u = 0; u < 4; ++u) {
                const int n = n4 + u;
                const int i = n / Wout, j = n - i * Wout;
                pv[u] = Xs[((i * STRIDE + kh) * Win + (j * STRIDE + kw)) * 32 + r];
            }
            *(float4*)(&Xc[k * Npad + n4]) = vv;
        }
    } else {
        for (int q = threadIdx.x; q < Ktot * PixOut; q += blockDim.x) {
            const int k = q / PixOut, n = q - k * PixOut;
            const int r = k / KK2, tt = k - r * KK2;
            const int kh = tt / KK, kw = tt - kh * KK;
            const int i = n / Wout, j = n - i * Wout;
            Xc[k * Npad + n] = Xs[((i * STRIDE + kh) * Win + (j * STRIDE + kw)) * 32 + r];
        }
    }
    __syncthreads();

    const int lane = threadIdx.x & 31;
    const int wave = threadIdx.x >> 5;
    const int half = lane >> 4;   // K-half select (lanes16-31 take K+2,K+3)
    const int lr   = lane & 15;

    for (int t = wave; t < Mt * Nt; t += 4) {
        const int m0 = (t / Nt) * 16;
        const int n0 = (t % Nt) * 16;

        // A: per-lane weight row (both lane halves cover M=0..15 at different K)
        int mA = m0 + lr; if (mA >= OCg) mA = OCg - 1;
        const float* wrow = W + (size_t)(g * OCg + mA) * Ktot;
        __builtin_prefetch(wrow, 0, 1);

        const int nB = n0 + lr;       // pad columns make clamping unnecessary
        v8f acc = {};
#pragma unroll 8
        for (int k0 = 0; k0 < Ktot; k0 += 4) {
            const int kb = k0 + (half << 1);
            const v2f av = *(const v2f*)(wrow + kb);       // global_load_b64
            v2f bvv;
            bvv.x = Xc[kb * Npad + nB];
            bvv.y = Xc[(kb + 1) * Npad + nB];
            acc = __builtin_amdgcn_wmma_f32_16x16x4_f32(false, av, false, bvv,
                                                        (short)0, acc, false, false);
        }

        // Epilogue. C layout: VGPR r -> M=m0+r (lanes0-15) / m0+r+8 (lanes16-31), N=n0+lr.
#pragma unroll
        for (int r = 0; r < 8; ++r) {
            const int m = m0 + r + (half << 3);
            if (m < OCg && nB < PixOut) {
                const int co = g * OCg + m;
                float v = scale * (acc[r] + Bias[co]);
                if constexpr (SIGCNT > 0) {
                    if (co < SIGCNT) v = 1.0f / (1.0f + __expf(-v));
                }
                const size_t oidx = NCHW ? ((size_t)co * PixOut + nB)
                                         : ((size_t)nB * OC_total + co);
                O[(size_t)b * outBS + oidx] = v;
            }
        }
    }
}

// ---------------------------------------------------------------------------
// Closed-form EM routing (exact collapse of the 3 reference iterations):
//   pose_out = 32*v  (stored directly by the votes kernel via scale=32)
//   S  = sum_c actconv[c];  C = sum_pd(beta_v - 0.5*log(30752*v^2)), v = P/32
//   ao = sigmoid(3*(beta_a - S*C))
// ---------------------------------------------------------------------------
template <int Npix, int Nout, int PC>
__global__ __launch_bounds__(128)
void em_kernel(const float* __restrict__ P, const float* __restrict__ AC,
               const float* __restrict__ bv, const float* __restrict__ ba,
               float* __restrict__ Aout,
               long long PBS, long long ACBS, long long ABS)
{
    const int b = blockIdx.x;
    const float* Pb = P + (size_t)b * PBS;
    const float* Ab = AC + (size_t)b * ACBS;
    float* Ob = Aout + (size_t)b * ABS;
    for (int idx = threadIdx.x; idx < Npix * Nout; idx += blockDim.x) {
        const int pix = idx / Nout;
        const int no  = idx - pix * Nout;
        float S = 0.0f;
#pragma unroll
        for (int c = 0; c < 32; ++c) S += Ab[pix * 32 + c];
        float C = 0.0f;
#pragma unroll
        for (int pd = 0; pd < 16; ++pd) {
            const float pm = Pb[(size_t)pix * PC + no * 16 + pd];  // = 32*v
            const float v  = pm * (1.0f / 32.0f);
            const float pv = 30752.0f * v * v;                     // 32*(31*v)^2
            C += bv[no * 16 + pd] - 0.5f * __logf(pv);
        }
        Ob[pix * Nout + no] = 1.0f / (1.0f + __expf(-3.0f * (ba[no] - S * C)));
    }
}

// ---------------------------------------------------------------------------
extern "C" void kernel_launch(void* const* d_in, const int* in_sizes, int n_in,
                              void* d_out, int out_size, void* d_ws, size_t ws_size,
                              hipStream_t stream)
{
    (void)in_sizes; (void)n_in; (void)out_size; (void)ws_size;
    const float* x   = (const float*)d_in[0];
    const float* c1w = (const float*)d_in[1];
    const float* c1b = (const float*)d_in[2];
    const float* pcw = (const float*)d_in[3];
    const float* pcb = (const float*)d_in[4];
    const float* t1w = (const float*)d_in[5];
    const float* t1b = (const float*)d_in[6];
    const float* a1w = (const float*)d_in[7];
    const float* a1b = (const float*)d_in[8];
    const float* bv1 = (const float*)d_in[9];
    const float* ba1 = (const float*)d_in[10];
    const float* t2w = (const float*)d_in[11];
    const float* t2b = (const float*)d_in[12];
    const float* a2w = (const float*)d_in[13];
    const float* a2b = (const float*)d_in[14];
    const float* bv2 = (const float*)d_in[15];
    const float* ba2 = (const float*)d_in[16];
    const float* t3w = (const float*)d_in[17];
    const float* t3b = (const float*)d_in[18];
    const float* a3w = (const float*)d_in[19];
    const float* a3b = (const float*)d_in[20];
    const float* bv3 = (const float*)d_in[21];
    const float* ba3 = (const float*)d_in[22];

    // Workspace layout (floats)
    float* ws  = (float*)d_ws;
    float* h   = ws;                 // 256*6272        = 1,605,632
    float* Y   = h   + 1605632;      // 256*544*196     = 27,295,744 (NCHW-flat pc out)
    float* AC1 = Y   + 27295744;     // 256*36*32       = 294,912
    float* A1  = AC1 + 294912;       // 256*36*32       = 294,912
    float* P1  = A1  + 294912;       // 256*36*512      = 4,718,592
    float* AC2 = P1  + 4718592;      // 256*16*32       = 131,072
    float* A2  = AC2 + 131072;       // 256*16*32       = 131,072
    float* P2  = A2  + 131072;       // 256*16*512      = 2,097,152
    float* V3  = P2  + 2097152;      // 256*160         = 40,960
    float* AC3 = V3  + 40960;        // 256*32          = 8,192

    // 1) conv1 + relu -> h (NHWC-flat)
    conv1_relu_kernel<<<256, 256, 0, stream>>>(x, c1w, c1b, h);

    // 2) primary-caps 1x1 conv: GEMM M=544 K=32 N=196 per image; sigmoid on ch<32;
    //    stored NCHW-flat so the reference's reshape is a pure reinterpretation.
    caps_conv_wmma_kernel<14,14,14,14,1,1,32,544,544,32,1>
        <<<dim3(256, 1), 128, 0, stream>>>(h, pcw, pcb, Y, 6272LL, 106624LL, 1.0f);

    // 3) layer1 votes: grouped 3x3 s2 conv on pose0 (=Y ch 32..543 reinterpreted);
    //    scale=32 stores pose1 = 32*v directly.
    caps_conv_wmma_kernel<14,14,6,6,3,2,512,32,512,0,0>
        <<<dim3(256, 16), 128, 0, stream>>>(Y + 6272, t1w, t1b, P1,
                                            106624LL, 18432LL, 32.0f);

    // 4) layer1 act conv (acts0 = sigmoid'd first 32 channels of Y, reinterpreted)
    caps_conv_wmma_kernel<14,14,6,6,3,2,32,32,32,0,0>
        <<<dim3(256, 1), 128, 0, stream>>>(Y, a1w, a1b, AC1,
                                           106624LL, 1152LL, 1.0f);

    // 5) layer1 EM (closed form) -> A1
    em_kernel<36,32,512><<<256, 128, 0, stream>>>(P1, AC1, bv1, ba1, A1,
                                                  18432LL, 1152LL, 1152LL);

    // 6) layer2 votes: grouped 3x3 s1 conv on pose1
    caps_conv_wmma_kernel<6,6,4,4,3,1,512,32,512,0,0>
        <<<dim3(256, 16), 128, 0, stream>>>(P1, t2w, t2b, P2,
                                            18432LL, 8192LL, 32.0f);

    // 7) layer2 act conv
    caps_conv_wmma_kernel<6,6,4,4,3,1,32,32,32,0,0>
        <<<dim3(256, 1), 128, 0, stream>>>(A1, a2w, a2b, AC2,
                                           1152LL, 512LL, 1.0f);

    // 8) layer2 EM -> A2
    em_kernel<16,32,512><<<256, 128, 0, stream>>>(P2, AC2, bv2, ba2, A2,
                                                  8192LL, 512LL, 512LL);

    // 9) layer3 votes: grouped 4x4 conv -> 1x1 spatial, 160 channels (= 32*v)
    caps_conv_wmma_kernel<4,4,1,1,4,1,512,10,160,0,0>
        <<<dim3(256, 16), 128, 0, stream>>>(P2, t3w, t3b, V3,
                                            8192LL, 160LL, 32.0f);

    // 10) layer3 act conv
    caps_conv_wmma_kernel<4,4,1,1,4,1,32,32,32,0,0>
        <<<dim3(256, 1), 128, 0, stream>>>(A2, a3w, a3b, AC3,
                                           512LL, 32LL, 1.0f);

    // 11) layer3 EM -> final activations (256 x 10) into d_out
    em_kernel<1,10,160><<<256, 128, 0, stream>>>(V3, AC3, bv3, ba3, (float*)d_out,
                                                 160LL, 32LL, 10LL);
}